// EfficientGCN_76287209111987
// MI455X (gfx1250) — compile-verified
//
#include <hip/hip_runtime.h>
#include <math.h>

// EfficientGCN preprocess for MI455X (gfx1250).
// x: (N=128, C=3, T=300, V=25, M=2) f32  ->  out: (N, 3, 6, T, V, M) f32
// Memory-bound: ~23 MB in, ~138 MB out. Strategy:
//   * block = (n, 30-frame tile); stage 3x(32 frames)x50 floats into LDS via
//     CDNA5 async global->LDS b128 DMA (ASYNCcnt), serve all reuse (center,
//     parent gather, t+1/t+2) from LDS.
//   * thread = (t, v) handling both m as float2; all 18 outputs per thread are
//     coalesced 8B non-temporal streaming stores (output is write-once).

typedef float f2 __attribute__((ext_vector_type(2)));

#define TB 30
#define PLANE 1600   // floats per staged channel plane: (TB+2)*50
#define PLANE2 800   // float2 units per plane

__global__ __launch_bounds__(256)
void egcn_preprocess(const float* __restrict__ x,
                     const int* __restrict__ conn,
                     float* __restrict__ out) {
  constexpr int T = 300;
  constexpr int TVM = 15000;          // T*V*M floats (contiguous per (n,c) plane)

  __shared__ __align__(16) float tile[3 * PLANE];
  __shared__ int s_conn[25];

  const int tid = threadIdx.x;
  const int bid = blockIdx.x;
  const int n   = bid / (T / TB);
  const int t0  = (bid % (T / TB)) * TB;

  if (tid < 25) s_conn[tid] = conn[tid];

  // ---- stage x[n, c, t0 .. t0+TB+1, :, :] into LDS with async DMA ----
  const int rem    = T - t0;
  const int frames = (rem < TB + 2) ? rem : (TB + 2);   // 32, or 30 on last tile (even)
  const int upp    = frames * 50 / 4;                   // b128 units per c-plane
  const int total_units = 3 * upp;
  const unsigned lds_base = (unsigned)(size_t)&tile[0]; // low 32 bits = LDS offset
  const unsigned long long src0 =
      (unsigned long long)(const void*)(x + (size_t)n * 3 * TVM + (size_t)t0 * 50);

  for (int u = tid; u < total_units; u += 256) {
    const int c = u / upp;
    const int j = u - c * upp;
    const unsigned long long g =
        src0 + (unsigned long long)c * (TVM * 4) + (unsigned long long)j * 16;
    const unsigned l = lds_base + (unsigned)(c * (PLANE * 4) + j * 16);
    asm volatile("global_load_async_to_lds_b128 %0, %1, off"
                 :: "v"(l), "v"(g) : "memory");
  }
#if __has_builtin(__builtin_amdgcn_s_wait_asynccnt)
  __builtin_amdgcn_s_wait_asynccnt(0);
#else
  asm volatile("s_wait_asynccnt 0" ::: "memory");
#endif
  __syncthreads();

  const f2* tp = (const f2*)tile;
  f2* o = (f2*)out + (size_t)n * 135000;   // 3*6*T*V*M / 2 float2 per sample

  const f2 zero = {0.0f, 0.0f};

  for (int e = tid; e < TB * 25; e += 256) {
    const int tl = e / 25;          // frame within tile
    const int v  = e - tl * 25;     // joint
    const int t  = t0 + tl;
    const int b0 = tl * 25;         // float2 index of frame start in plane
    const int pv = s_conn[v];
    const bool hasVel = (t <= T - 3);

    f2 self[3], rel[3], d1[3], d2[3], bv[3];
#pragma unroll
    for (int c = 0; c < 3; ++c) {
      const f2* P = tp + c * PLANE2;
      const f2 s  = P[b0 + v];        // x[n,c,t,v,:]
      const f2 ce = P[b0 + 1];        // center joint (v=1)
      const f2 pa = P[b0 + pv];       // parent joint
      const f2 x1 = P[b0 + 25 + v];   // x[.., t+1, ..] (in-tile; discarded if !hasVel)
      const f2 x2 = P[b0 + 50 + v];   // x[.., t+2, ..]
      self[c] = s;
      rel[c]  = s - ce;
      d1[c]   = hasVel ? (f2)(x1 - s) : zero;
      d2[c]   = hasVel ? (f2)(x2 - s) : zero;
      bv[c]   = s - pa;
    }

    const f2 ss = bv[0] * bv[0] + bv[1] * bv[1] + bv[2] * bv[2];
    f2 len;
    len.x = sqrtf(ss.x) + 1e-4f;
    len.y = sqrtf(ss.y) + 1e-4f;

    const int r = t * 25 + v;        // float2 index within a (branch,channel) plane
#pragma unroll
    for (int c = 0; c < 3; ++c) {
      f2 ang;
      ang.x = acosf(bv[c].x / len.x);
      ang.y = acosf(bv[c].y / len.y);
      __builtin_nontemporal_store(self[c], o + (0 * 6 + c)     * 7500 + r); // joint: x
      __builtin_nontemporal_store(rel[c],  o + (0 * 6 + 3 + c) * 7500 + r); // joint: x-center
      __builtin_nontemporal_store(d1[c],   o + (1 * 6 + c)     * 7500 + r); // vel: 1-step
      __builtin_nontemporal_store(d2[c],   o + (1 * 6 + 3 + c) * 7500 + r); // vel: 2-step
      __builtin_nontemporal_store(bv[c],   o + (2 * 6 + c)     * 7500 + r); // bone vec
      __builtin_nontemporal_store(ang,     o + (2 * 6 + 3 + c) * 7500 + r); // bone angle
    }
  }
}

extern "C" void kernel_launch(void* const* d_in, const int* in_sizes, int n_in,
                              void* d_out, int out_size, void* d_ws, size_t ws_size,
                              hipStream_t stream) {
  const float* x   = (const float*)d_in[0];
  const int* conn  = (const int*)d_in[1];
  float* out       = (float*)d_out;
  (void)in_sizes; (void)n_in; (void)out_size; (void)d_ws; (void)ws_size;

  const int blocks = 128 * (300 / TB);   // one block per (sample, 30-frame tile)
  egcn_preprocess<<<blocks, 256, 0, stream>>>(x, conn, out);
}